// AttentionHead_10230612099561
// MI455X (gfx1250) — compile-verified
//
#include <hip/hip_runtime.h>

typedef __attribute__((ext_vector_type(2))) float v2f;
typedef __attribute__((ext_vector_type(8))) float v8f;

#define EMBED 1024
#define SEQ   2048
#define HEAD  64
#define BATCH 4
#define SCALE 0.03125f   // EMBED^-0.5

__device__ __forceinline__ v8f wmma_f32(v2f a, v2f b, v8f c) {
    // V_WMMA_F32_16X16X4_F32 : D = A(16x4) * B(4x16) + C(16x16), all fp32
    return __builtin_amdgcn_wmma_f32_16x16x4_f32(false, a, false, b, (short)0, c,
                                                 false, false);
}

// ---------------------------------------------------------------------------
// Kernel 1: Q/K/V projection.  One wave computes one 16x64 output tile of one
// of the three matrices.  X tile is the WMMA A operand (lane = row, half-wave
// splits K), W chunk is the B operand (lane = col, half-wave splits K).
// ---------------------------------------------------------------------------
__global__ __launch_bounds__(256) void qkv_proj_kernel(
    const float* __restrict__ tokens,
    const float* __restrict__ Wq,
    const float* __restrict__ Wk,
    const float* __restrict__ Wv,
    float* __restrict__ Qo, float* __restrict__ Ko, float* __restrict__ Vo)
{
    const int lane = threadIdx.x & 31;
    const int wid  = blockIdx.x * (blockDim.x >> 5) + (threadIdx.x >> 5);
    const int mat  = wid % 3;
    const int rt   = wid / 3;              // 16-row tile index in [0, 512)
    const int r    = lane & 15;
    const int hi   = lane >> 4;

    const float* W   = (mat == 0) ? Wq : (mat == 1) ? Wk : Wv;
    float*       Out = ((mat == 0) ? Qo : (mat == 1) ? Ko : Vo)
                       + (size_t)rt * 16 * HEAD;
    const float* X   = tokens + (size_t)rt * 16 * EMBED + (size_t)r * EMBED;

    v8f acc0 = {}, acc1 = {}, acc2 = {}, acc3 = {};
    for (int t = 0; t < EMBED / 4; ++t) {
        const int k0 = 4 * t + 2 * hi;
        v2f a;  a.x = X[k0];       a.y = X[k0 + 1];
        const float* Wr0 = W + (size_t)k0 * HEAD + r;
        const float* Wr1 = Wr0 + HEAD;
        v2f b0; b0.x = Wr0[0];  b0.y = Wr1[0];
        v2f b1; b1.x = Wr0[16]; b1.y = Wr1[16];
        v2f b2; b2.x = Wr0[32]; b2.y = Wr1[32];
        v2f b3; b3.x = Wr0[48]; b3.y = Wr1[48];
        acc0 = wmma_f32(a, b0, acc0);
        acc1 = wmma_f32(a, b1, acc1);
        acc2 = wmma_f32(a, b2, acc2);
        acc3 = wmma_f32(a, b3, acc3);
    }
    // D layout: lane, reg i -> row (i + 8*hi), col r of each 16-col tile
    #pragma unroll
    for (int i = 0; i < 8; ++i) {
        const int row = i + 8 * hi;
        Out[row * HEAD +  0 + r] = acc0[i];
        Out[row * HEAD + 16 + r] = acc1[i];
        Out[row * HEAD + 32 + r] = acc2[i];
        Out[row * HEAD + 48 + r] = acc3[i];
    }
}

// ---------------------------------------------------------------------------
// Kernel 2: causal flash attention.  One wave owns one (batch, 16-query tile).
// We compute S^T = K * Q^T so that query index = lane (softmax is lane-local),
// and O^T = V^T * P^T so the online-softmax rescale is a per-lane scalar mul.
// ---------------------------------------------------------------------------
__global__ __launch_bounds__(256) void flash_attn_kernel(
    const float* __restrict__ Q, const float* __restrict__ K,
    const float* __restrict__ V, float* __restrict__ O)
{
    const int lane = threadIdx.x & 31;
    const int wid  = blockIdx.x * (blockDim.x >> 5) + (threadIdx.x >> 5);
    const int b    = wid >> 7;             // 128 q-tiles per batch
    const int qi   = wid & 127;
    const int r    = lane & 15;            // query (and key-row) within tile
    const int hi   = lane >> 4;

    const size_t base = (size_t)b * SEQ * HEAD;
    const float* Qp   = Q + base + (size_t)(qi * 16 + r) * HEAD;

    // Q tile preloaded in B-operand layout: lane holds Q[r][4t+2hi+{0,1}]
    float qreg[32];
    #pragma unroll
    for (int t = 0; t < 16; ++t) {
        const int c  = 4 * t + 2 * hi;
        qreg[2 * t]     = Qp[c];
        qreg[2 * t + 1] = Qp[c + 1];
    }

    float m = -3.0e38f, l = 0.0f;
    v8f o[4] = {};                         // O^T accum: o[mt][i] = O[q=r][16mt+8hi+i]

    for (int ki = 0; ki <= qi; ++ki) {
        const float* Kp = K + base + (size_t)(ki * 16 + r) * HEAD;

        // S^T(16x16) = K_tile(16x64) * Q_tile^T(64x16)
        v8f s = {};
        #pragma unroll
        for (int t = 0; t < 16; ++t) {
            const int c = 4 * t + 2 * hi;
            v2f a;  a.x  = Kp[c];          a.y  = Kp[c + 1];
            v2f bb; bb.x = qreg[2 * t];    bb.y = qreg[2 * t + 1];
            s = wmma_f32(a, bb, s);
        }

        // s[i] = S^T[key = i+8hi][query = r]; scale + causal mask
        float sv[8];
        #pragma unroll
        for (int i = 0; i < 8; ++i) sv[i] = s[i] * SCALE;
        if (ki == qi) {
            #pragma unroll
            for (int i = 0; i < 8; ++i)
                if ((i + 8 * hi) > r) sv[i] = -1.0e9f;
        }

        // online softmax (query-local; combine both key halves via xor-16)
        float mt = sv[0];
        #pragma unroll
        for (int i = 1; i < 8; ++i) mt = fmaxf(mt, sv[i]);
        mt = fmaxf(mt, __shfl_xor(mt, 16, 32));
        const float mnew = fmaxf(m, mt);
        const float corr = __expf(m - mnew);

        float p[8];
        float rs = 0.0f;
        #pragma unroll
        for (int i = 0; i < 8; ++i) { p[i] = __expf(sv[i] - mnew); rs += p[i]; }
        rs += __shfl_xor(rs, 16, 32);
        l = l * corr + rs;
        m = mnew;

        #pragma unroll
        for (int mtI = 0; mtI < 4; ++mtI)
            #pragma unroll
            for (int i = 0; i < 8; ++i) o[mtI][i] *= corr;

        // O^T += V^T(64x16) * P^T(16x16)
        const float* Vp = V + base + (size_t)(ki * 16) * HEAD;
        #pragma unroll
        for (int kt = 0; kt < 4; ++kt) {
            const int u = 4 * (kt & 1);
            float b0, b1;
            if (kt < 2) {   // keys 4kt..4kt+3 live in the low half's regs
                const float r0 = __shfl_xor(p[u + 2], 16, 32);
                const float r1 = __shfl_xor(p[u + 3], 16, 32);
                b0 = hi ? r0 : p[u + 0];
                b1 = hi ? r1 : p[u + 1];
            } else {        // keys live in the high half's regs
                const float r0 = __shfl_xor(p[u + 0], 16, 32);
                const float r1 = __shfl_xor(p[u + 1], 16, 32);
                b0 = hi ? p[u + 2] : r0;
                b1 = hi ? p[u + 3] : r1;
            }
            v2f bb; bb.x = b0; bb.y = b1;

            const int key = 4 * kt + 2 * hi;
            const float* Vk0 = Vp + (size_t)key * HEAD + r;
            const float* Vk1 = Vk0 + HEAD;
            v2f a;
            a.x = Vk0[0];  a.y = Vk1[0];  o[0] = wmma_f32(a, bb, o[0]);
            a.x = Vk0[16]; a.y = Vk1[16]; o[1] = wmma_f32(a, bb, o[1]);
            a.x = Vk0[32]; a.y = Vk1[32]; o[2] = wmma_f32(a, bb, o[2]);
            a.x = Vk0[48]; a.y = Vk1[48]; o[3] = wmma_f32(a, bb, o[3]);
        }
    }

    const float inv_l = 1.0f / l;
    // o[mt][i] = O[query r][16mt + 8hi + i] -> 8 contiguous floats per lane/mt
    float* Op = O + base + (size_t)(qi * 16 + r) * HEAD + 8 * hi;
    #pragma unroll
    for (int mtI = 0; mtI < 4; ++mtI) {
        float* dst = Op + 16 * mtI;
        #pragma unroll
        for (int i = 0; i < 8; ++i) dst[i] = o[mtI][i] * inv_l;
    }
}

// ---------------------------------------------------------------------------
extern "C" void kernel_launch(void* const* d_in, const int* in_sizes, int n_in,
                              void* d_out, int out_size, void* d_ws, size_t ws_size,
                              hipStream_t stream) {
    const float* tokens = (const float*)d_in[0];
    const float* Wq     = (const float*)d_in[1];
    const float* Wk     = (const float*)d_in[2];
    const float* Wv     = (const float*)d_in[3];
    float* out = (float*)d_out;

    const size_t qkv_elems = (size_t)BATCH * SEQ * HEAD;   // 512K floats = 2 MB
    float* Qw = (float*)d_ws;
    float* Kw = Qw + qkv_elems;
    float* Vw = Kw + qkv_elems;

    // Phase 1: 512 row-tiles * 3 matrices = 1536 waves, 8 waves/block
    const int waves1 = (BATCH * SEQ / 16) * 3;
    qkv_proj_kernel<<<waves1 / 8, 256, 0, stream>>>(tokens, Wq, Wk, Wv, Qw, Kw, Vw);

    // Phase 2: 4 batches * 128 q-tiles = 512 waves, 8 waves/block
    const int waves2 = BATCH * (SEQ / 16);
    flash_attn_kernel<<<waves2 / 8, 256, 0, stream>>>(Qw, Kw, Vw, out);
}